// DRGCN_61332132987262
// MI455X (gfx1250) — compile-verified
//
#include <hip/hip_runtime.h>
#include <hip/hip_bf16.h>

typedef __attribute__((ext_vector_type(2))) float v2f;
typedef __attribute__((ext_vector_type(8))) float v8f;

static constexpr int N_NODES = 50000;
static constexpr int N_EDGES = 800000;
static constexpr int D       = 128;
static constexpr float INV_TEMP = 0.1f;
// W rows are N_BASES*SUBMAT*SUBMAT = 32*4*4 = 512 floats

// gfx1250 async copy: global -> LDS, 16 bytes per lane, tracked by ASYNCcnt.
__device__ __forceinline__ void async_ld_b128(unsigned int lds_addr, const void* gaddr)
{
    asm volatile("global_load_async_to_lds_b128 %0, %1, off"
                 :: "v"(lds_addr), "v"(gaddr)
                 : "memory");
}
__device__ __forceinline__ void wait_asynccnt0()
{
    asm volatile("s_wait_asynccnt 0" ::: "memory");
}

// ---------------------------------------------------------------------------
// Dense kernel: dense[n] = A1[n] @ W1 + decay[n] * (A2[n] @ W2)
// One wave per 16-row stripe (4 waves/block). W1/W2 staged into LDS in two
// 64KB phases via GLOBAL_LOAD_ASYNC_TO_LDS_B128; fp32 WMMA 16x16x4, K=128.
// ---------------------------------------------------------------------------
__global__ void __launch_bounds__(128)
drgcn_dense_wmma(const float* __restrict__ A1,
                 const float* __restrict__ A2,
                 const float* __restrict__ W1,
                 const float* __restrict__ W2,
                 const float* __restrict__ tdiff,
                 float* __restrict__ outD)
{
    __shared__ float smem[16384];              // 64KB: [0..8191]=W1 half, [8192..]=W2 half

    const int lane   = threadIdx.x & 31;
    const int wave   = threadIdx.x >> 5;
    const int stripe = blockIdx.x * 4 + wave;
    const int m0     = stripe * 16;
    const bool active = (m0 < N_NODES);        // wave-uniform; inactive waves still barrier

    const int mlane = active ? (m0 + (lane & 15)) : (lane & 15);
    const int khalf = (lane >> 4) << 1;        // 0 for lanes 0-15, 2 for lanes 16-31
    const int ncol  = (lane & 15);             // column within N-tile (B/C layout)
    const float decay = __expf(-tdiff[mlane] * INV_TEMP);

    const float* __restrict__ a1row = A1 + (size_t)mlane * D;
    const float* __restrict__ a2row = A2 + (size_t)mlane * D;

    const unsigned int lbase = (unsigned int)(unsigned long long)&smem[0]; // LDS aperture: low 32 bits = offset

    v8f acc[8];
#pragma unroll
    for (int t = 0; t < 8; ++t) acc[t] = v8f{0.f,0.f,0.f,0.f,0.f,0.f,0.f,0.f};

    for (int phase = 0; phase < 2; ++phase) {
        const int kbase = phase * 64;
        if (phase) __syncthreads();            // previous phase fully consumed

        // Stage rows [kbase, kbase+64) of W1 and W2 (8192 floats each) into LDS.
        const float* g1 = W1 + (size_t)kbase * D;
        const float* g2 = W2 + (size_t)kbase * D;
#pragma unroll
        for (int i = 0; i < 16; ++i) {
            const int f4 = threadIdx.x + i * 128;            // float4 index 0..2047
            async_ld_b128(lbase + (unsigned)f4 * 16u,          g1 + f4 * 4);
            async_ld_b128(lbase + 32768u + (unsigned)f4 * 16u, g2 + f4 * 4);
        }
        wait_asynccnt0();
        __syncthreads();

        const float* s1 = smem;                // rows kbase..kbase+63 of W1
        const float* s2 = smem + 8192;         // rows kbase..kbase+63 of W2

        for (int k0 = kbase; k0 < kbase + 64; k0 += 4) {
            // A fragments (16x4 f32): vgpr v holds K = k0 + khalf + v
            v2f a1; a1.x = a1row[k0 + khalf];  a1.y = a1row[k0 + khalf + 1];
            v2f a2; a2.x = a2row[k0 + khalf] * decay;
                    a2.y = a2row[k0 + khalf + 1] * decay;

            const int krel = k0 - kbase + khalf;
            const float* w1r0 = s1 + krel * D;
            const float* w2r0 = s2 + krel * D;
#pragma unroll
            for (int nt = 0; nt < 8; ++nt) {
                const int n = nt * 16 + ncol;
                // B fragments (4x16 f32): vgpr v holds row K = k0+khalf+v, col n
                v2f b1; b1.x = w1r0[n]; b1.y = w1r0[D + n];
                acc[nt] = __builtin_amdgcn_wmma_f32_16x16x4_f32(
                    false, a1, false, b1, (short)0, acc[nt], false, false);
                v2f b2; b2.x = w2r0[n]; b2.y = w2r0[D + n];
                acc[nt] = __builtin_amdgcn_wmma_f32_16x16x4_f32(
                    false, a2, false, b2, (short)0, acc[nt], false, false);
            }
        }
    }

    if (active) {
        // C layout: vgpr v -> row m0 + v + 8*(lane/16), col nt*16 + (lane%16)
        const int rbase = m0 + ((lane >> 4) << 3);
#pragma unroll
        for (int nt = 0; nt < 8; ++nt) {
            const int col = nt * 16 + ncol;
#pragma unroll
            for (int v = 0; v < 8; ++v) {
                outD[(size_t)(rbase + v) * D + col] = acc[nt][v];
            }
        }
    }
}

// ---------------------------------------------------------------------------
// Edge kernel: one wave per edge, lane = base-block b (0..31).
// msg[b*4+o] = sum_i h[src][b*4+i] * W[type][b*16 + i*4 + o], * edge_norm,
// atomically scattered into agg[dst] with hardware f32 atomics.
// ---------------------------------------------------------------------------
__global__ void __launch_bounds__(256)
drgcn_edge_msg(const float* __restrict__ h,
               const int*   __restrict__ esrc,
               const int*   __restrict__ edst,
               const int*   __restrict__ etype,
               const float* __restrict__ enorm,
               const float* __restrict__ W,
               float*       __restrict__ agg)
{
    const int lane = threadIdx.x & 31;
    const int e    = blockIdx.x * 8 + (threadIdx.x >> 5);
    if (e >= N_EDGES) return;                  // wave-uniform

    const int   src  = esrc[e];
    const int   dst  = edst[e];
    const int   t    = etype[e];
    const float norm = enorm[e];

    const float4 hs = *(const float4*)(h + (size_t)src * D + lane * 4);
    const float* wb = W + (size_t)t * 512 + lane * 16;
    const float4 w0 = *(const float4*)(wb);       // i = 0, o = 0..3
    const float4 w1 = *(const float4*)(wb + 4);   // i = 1
    const float4 w2 = *(const float4*)(wb + 8);   // i = 2
    const float4 w3 = *(const float4*)(wb + 12);  // i = 3

    const float o0 = hs.x*w0.x + hs.y*w1.x + hs.z*w2.x + hs.w*w3.x;
    const float o1 = hs.x*w0.y + hs.y*w1.y + hs.z*w2.y + hs.w*w3.y;
    const float o2 = hs.x*w0.z + hs.y*w1.z + hs.z*w2.z + hs.w*w3.z;
    const float o3 = hs.x*w0.w + hs.y*w1.w + hs.z*w2.w + hs.w*w3.w;

    float* ap = agg + (size_t)dst * D + lane * 4;
    __hip_atomic_fetch_add(ap + 0, o0 * norm, __ATOMIC_RELAXED, __HIP_MEMORY_SCOPE_AGENT);
    __hip_atomic_fetch_add(ap + 1, o1 * norm, __ATOMIC_RELAXED, __HIP_MEMORY_SCOPE_AGENT);
    __hip_atomic_fetch_add(ap + 2, o2 * norm, __ATOMIC_RELAXED, __HIP_MEMORY_SCOPE_AGENT);
    __hip_atomic_fetch_add(ap + 3, o3 * norm, __ATOMIC_RELAXED, __HIP_MEMORY_SCOPE_AGENT);
}

// ---------------------------------------------------------------------------
// Combine: out = agg * node_norm + dense   (optional ReLU)
// ---------------------------------------------------------------------------
__global__ void __launch_bounds__(256)
drgcn_combine(const float* __restrict__ agg,
              const float* __restrict__ dense,
              const float* __restrict__ node_norm,
              float* __restrict__ out,
              int relu)
{
    const int idx = blockIdx.x * blockDim.x + threadIdx.x;   // float4 index
    if (idx >= N_NODES * (D / 4)) return;
    const int row = idx >> 5;                                // 32 float4 groups / row
    const float nn = node_norm[row];
    const float4 a = ((const float4*)agg)[idx];
    const float4 d = ((const float4*)dense)[idx];
    float4 r;
    r.x = a.x * nn + d.x;
    r.y = a.y * nn + d.y;
    r.z = a.z * nn + d.z;
    r.w = a.w * nn + d.w;
    if (relu) {
        r.x = fmaxf(r.x, 0.f); r.y = fmaxf(r.y, 0.f);
        r.z = fmaxf(r.z, 0.f); r.w = fmaxf(r.w, 0.f);
    }
    ((float4*)out)[idx] = r;
}

// ---------------------------------------------------------------------------
extern "C" void kernel_launch(void* const* d_in, const int* in_sizes, int n_in,
                              void* d_out, int out_size, void* d_ws, size_t ws_size,
                              hipStream_t stream)
{
    const float* h           = (const float*)d_in[0];
    const float* first_prev  = (const float*)d_in[1];
    const float* second_prev = (const float*)d_in[2];
    const float* time_diff   = (const float*)d_in[3];
    const int*   edge_src    = (const int*)d_in[4];
    const int*   edge_dst    = (const int*)d_in[5];
    const int*   edge_type   = (const int*)d_in[6];
    const float* edge_norm   = (const float*)d_in[7];
    const float* node_norm   = (const float*)d_in[8];
    const float* W1          = (const float*)d_in[9];
    const float* W2          = (const float*)d_in[10];
    const float* loop_w1     = (const float*)d_in[11];
    const float* loop_w2     = (const float*)d_in[12];
    const float* time_w1     = (const float*)d_in[13];
    const float* time_w2     = (const float*)d_in[14];

    const size_t nd = (size_t)N_NODES * D;
    float* agg   = (float*)d_ws;
    float* dense = agg + nd;

    float* h1 = (float*)d_out;
    float* h2 = h1 + nd;

    const int denseBlocks   = ( (N_NODES / 16) + 3 ) / 4;        // 4 waves/block
    const int edgeBlocks    = (N_EDGES + 7) / 8;                 // 8 edges/block
    const int combineBlocks = (N_NODES * (D / 4) + 255) / 256;

    // ---- Layer 1 (no activation) ----
    hipMemsetAsync(agg, 0, nd * sizeof(float), stream);
    drgcn_dense_wmma<<<denseBlocks, 128, 0, stream>>>(
        h, first_prev, loop_w1, time_w1, time_diff, dense);
    drgcn_edge_msg<<<edgeBlocks, 256, 0, stream>>>(
        h, edge_src, edge_dst, edge_type, edge_norm, W1, agg);
    drgcn_combine<<<combineBlocks, 256, 0, stream>>>(
        agg, dense, node_norm, h1, /*relu=*/0);

    // ---- Layer 2 (ReLU), input is h1 ----
    hipMemsetAsync(agg, 0, nd * sizeof(float), stream);
    drgcn_dense_wmma<<<denseBlocks, 128, 0, stream>>>(
        h1, second_prev, loop_w2, time_w2, time_diff, dense);
    drgcn_edge_msg<<<edgeBlocks, 256, 0, stream>>>(
        h1, edge_src, edge_dst, edge_type, edge_norm, W2, agg);
    drgcn_combine<<<combineBlocks, 256, 0, stream>>>(
        agg, dense, node_norm, h2, /*relu=*/1);
}